// ViscoelasticWave3D_44727789420616
// MI455X (gfx1250) — compile-verified
//
#include <hip/hip_runtime.h>
#include <hip/hip_bf16.h>

// ---------------- static problem config (matches reference) ----------------
constexpr int NX = 128, NY = 128, NZ = 64;
constexpr int B  = 2,  T  = 12;
constexpr int SX = NY * NZ;      // x stride = 8192
constexpr int SY = NZ;           // y stride = 64
constexpr int N3 = NX * NY * NZ; // 1,048,576
constexpr int N4 = B * N3;       // 2,097,152 (one field)

constexpr float DT    = 1e-3f;
constexpr float INV   = 0.1f;            // 1/dx = 1/dy = 1/dz = 0.1
constexpr float BULK  = 1.0f - 0.01f * 1e-3f;
constexpr float INV2SXY = 1.0f / (2.0f * 20.0f * 20.0f);
constexpr float INV2SZ  = 1.0f / (2.0f * 10.0f * 10.0f);
constexpr float SRC_DEPTH = 320.0f;
constexpr float PHYS_EPS  = 1e-8f;

typedef float f4  __attribute__((ext_vector_type(4)));
typedef float v2f __attribute__((ext_vector_type(2)));
typedef float v8f __attribute__((ext_vector_type(8)));

__device__ __forceinline__ f4 ld4(const float* p) { return *(const f4*)p; }
__device__ __forceinline__ void st4(float* p, f4 v) { *(f4*)p = v; }
__device__ __forceinline__ f4 splat(float s) { f4 r; r.x = s; r.y = s; r.z = s; r.w = s; return r; }

// forward diff along x or y (stride), zero at far edge (uniform predicate)
__device__ __forceinline__ f4 fd_s(const float* p, size_t idx, f4 c, bool e, int stride) {
  if (!e) return splat(0.0f);
  return (ld4(p + idx + stride) - c) * INV;
}
// backward diff along x or y, zero at near edge
__device__ __forceinline__ f4 bd_s(const float* p, size_t idx, f4 c, bool e, int stride) {
  if (!e) return splat(0.0f);
  return (c - ld4(p + idx - stride)) * INV;
}
// forward diff along z within a quad; last element needs p[idx+4] unless k+4==NZ
__device__ __forceinline__ f4 fd_z(const float* p, size_t idx, f4 c, bool e) {
  f4 r;
  r.x = c.y - c.x; r.y = c.z - c.y; r.z = c.w - c.z;
  float nxt = 0.0f;
  if (e) nxt = p[idx + 4];
  r.w = e ? (nxt - c.w) : 0.0f;
  return r * INV;
}
// backward diff along z; first element needs p[idx-1] unless k==0
__device__ __forceinline__ f4 bd_z(const float* p, size_t idx, f4 c, int k) {
  f4 r;
  float prev = 0.0f;
  if (k > 0) prev = p[idx - 1];
  r.x = (k > 0) ? (c.x - prev) : 0.0f;
  r.y = c.y - c.x; r.z = c.z - c.y; r.w = c.w - c.z;
  return r * INV;
}

// ---------------- source profile precompute (tiny) ----------------
__global__ __launch_bounds__(256) void profiles_kernel(
    const float* __restrict__ traj, const float* __restrict__ gx,
    const float* __restrict__ gy,   const float* __restrict__ gz,
    float* __restrict__ px, float* __restrict__ py, float* __restrict__ pz)
{
  const int TOT = T * B * NX; // 3072
  for (int e = blockIdx.x * blockDim.x + threadIdx.x; e < 2 * TOT + NZ;
       e += blockDim.x * gridDim.x) {
    if (e < TOT) {
      int t = e / (B * NX); int rem = e % (B * NX); int b = rem / NX; int i = rem % NX;
      float xi = gx[(size_t)i * NY * NZ];
      float xt = traj[(b * T + t) * 2 + 0];
      float d = xi - xt;
      px[e] = __expf(-d * d * INV2SXY);
    } else if (e < 2 * TOT) {
      int e2 = e - TOT;
      int t = e2 / (B * NX); int rem = e2 % (B * NX); int b = rem / NX; int j = rem % NX;
      float yj = gy[(size_t)j * NZ];
      float yt = traj[(b * T + t) * 2 + 1];
      float d = yj - yt;
      py[e2] = __expf(-d * d * INV2SXY);
    } else {
      int k = e - 2 * TOT;
      float d = gz[k] - SRC_DEPTH;
      pz[k] = __expf(-d * d * INV2SZ);
    }
  }
}

// ---------------- time-invariant material coefficients (once per call) ------
// c_l2m = DT*(lam+2mu); c_lam = DT*lam; c_me2 = 2*DT*(mu+eta);
// dt_rho = DT/rho (kills f32 division in the hot loop); dampb = damping*BULK.
__global__ __launch_bounds__(256) void matcoef_kernel(
    const float* __restrict__ mu, const float* __restrict__ lam,
    const float* __restrict__ eta, const float* __restrict__ rho,
    const float* __restrict__ damp,
    float* __restrict__ c_l2m, float* __restrict__ c_lam, float* __restrict__ c_me2,
    float* __restrict__ dt_rho, float* __restrict__ dampb)
{
  size_t q = ((size_t)blockIdx.x * blockDim.x + threadIdx.x) * 4;
  if (q >= (size_t)N3) return;
  f4 mu4 = ld4(mu + q), lam4 = ld4(lam + q), eta4 = ld4(eta + q);
  f4 rho4 = ld4(rho + q), dm4 = ld4(damp + q);
  st4(c_l2m + q, DT * (lam4 + 2.0f * mu4));
  st4(c_lam + q, DT * lam4);
  st4(c_me2 + q, (2.0f * DT) * (mu4 + eta4));
  st4(dt_rho + q, splat(DT) / rho4);
  st4(dampb + q, dm4 * BULK);
}

// ---------------- rank-1 outer product px ⊗ py via WMMA ----------------
// srctab[tb][i][j] = px[tb][i] * py[tb][j];  one wave computes one 16x16 tile.
__global__ __launch_bounds__(256) void srcxy_wmma(
    const float* __restrict__ px, const float* __restrict__ py,
    float* __restrict__ srctab)
{
  const int flat = blockIdx.x * blockDim.x + threadIdx.x;
  const int wave = flat >> 5;
  const int lane = flat & 31;
  const int tb   = wave >> 6;   // (t*B+b) in [0,24)
  const int tile = wave & 63;   // 8x8 tiles over 128x128
  const int ti = tile >> 3, tj = tile & 7;
  const int l15 = lane & 15;

  // A (16x4): VGPR0 = {lanes0-15: K=0, lanes16-31: K=2}; VGPR1 = {K=1, K=3}
  v2f a;  a.x  = (lane < 16) ? px[tb * NX + ti * 16 + l15] : 0.0f;  a.y  = 0.0f;
  v2f bm; bm.x = (lane < 16) ? py[tb * NY + tj * 16 + l15] : 0.0f;  bm.y = 0.0f;
  v8f c = {};
  v8f d = __builtin_amdgcn_wmma_f32_16x16x4_f32(false, a, false, bm,
                                                (short)0, c, false, false);

  // D layout: VGPR r -> lanes0-15: M=r, lanes16-31: M=r+8; N = lane%16
  const int row0 = ti * 16 + ((lane < 16) ? 0 : 8);
  const int col  = tj * 16 + l15;
  float* outp = srctab + (size_t)tb * NX * NY + (size_t)row0 * NY + col;
#pragma unroll
  for (int r = 0; r < 8; ++r) outp[(size_t)r * NY] = d[r];
}

// ---------------- stress update: s += C : grad_fwd(v)  (coeffs pre-scaled) ---
__global__ __launch_bounds__(256) void stress_step(
    const float* __restrict__ vx, const float* __restrict__ vy, const float* __restrict__ vz,
    float* __restrict__ sxx, float* __restrict__ syy, float* __restrict__ szz,
    float* __restrict__ sxy, float* __restrict__ sxz, float* __restrict__ syz,
    const float* __restrict__ c_l2m, const float* __restrict__ c_lam,
    const float* __restrict__ c_me2)
{
  const int k = threadIdx.x * 4;
  const int j = blockIdx.x * 16 + threadIdx.y;
  const int i = blockIdx.y;
  const int b = blockIdx.z;
  const size_t m   = ((size_t)i * NY + j) * NZ + k;
  const size_t idx = (size_t)b * N3 + m;
  const bool xe = (i + 1 < NX), ye = (j + 1 < NY), ze = (k + 4 < NZ);

  if (xe) __builtin_prefetch(vx + idx + 2 * (size_t)SX, 0, 1);

  f4 vx0 = ld4(vx + idx), vy0 = ld4(vy + idx), vz0 = ld4(vz + idx);

  f4 exx = fd_s(vx, idx, vx0, xe, SX);
  f4 eyy = fd_s(vy, idx, vy0, ye, SY);
  f4 ezz = fd_z(vz, idx, vz0, ze);
  f4 exy = 0.5f * (fd_s(vx, idx, vx0, ye, SY) + fd_s(vy, idx, vy0, xe, SX));
  f4 exz = 0.5f * (fd_z(vx, idx, vx0, ze)     + fd_s(vz, idx, vz0, xe, SX));
  f4 eyz = 0.5f * (fd_z(vy, idx, vy0, ze)     + fd_s(vz, idx, vz0, ye, SY));

  f4 l2m  = ld4(c_l2m + m), lamc = ld4(c_lam + m), me2 = ld4(c_me2 + m);

  st4(sxx + idx, ld4(sxx + idx) + l2m * exx + lamc * (eyy + ezz));
  st4(syy + idx, ld4(syy + idx) + l2m * eyy + lamc * (exx + ezz));
  st4(szz + idx, ld4(szz + idx) + l2m * ezz + lamc * (exx + eyy));
  st4(sxy + idx, ld4(sxy + idx) + me2 * exy);
  st4(sxz + idx, ld4(sxz + idx) + me2 * exz);
  st4(syz + idx, ld4(syz + idx) + me2 * eyz);
}

// ---------------- velocity update: v = (v + div_bwd(s)*dt_rho) * dampb -------
__global__ __launch_bounds__(256) void velocity_step(
    float* __restrict__ vx, float* __restrict__ vy, float* __restrict__ vz,
    const float* __restrict__ sxx, const float* __restrict__ syy, const float* __restrict__ szz,
    const float* __restrict__ sxy, const float* __restrict__ sxz, const float* __restrict__ syz,
    const float* __restrict__ dt_rho, const float* __restrict__ dampb,
    const float* __restrict__ srcxy, const float* __restrict__ pz)
{
  const int k = threadIdx.x * 4;
  const int j = blockIdx.x * 16 + threadIdx.y;
  const int i = blockIdx.y;
  const int b = blockIdx.z;
  const size_t m   = ((size_t)i * NY + j) * NZ + k;
  const size_t idx = (size_t)b * N3 + m;
  const bool xe = (i > 0), ye = (j > 0);

  if (i + 1 < NX) __builtin_prefetch(sxx + idx + 2 * (size_t)SX, 0, 1);

  f4 sxx0 = ld4(sxx + idx), syy0 = ld4(syy + idx), szz0 = ld4(szz + idx);
  f4 sxy0 = ld4(sxy + idx), sxz0 = ld4(sxz + idx), syz0 = ld4(syz + idx);

  f4 div_x = bd_s(sxx, idx, sxx0, xe, SX) + bd_s(sxy, idx, sxy0, ye, SY) + bd_z(sxz, idx, sxz0, k);
  f4 div_y = bd_s(sxy, idx, sxy0, xe, SX) + bd_s(syy, idx, syy0, ye, SY) + bd_z(syz, idx, syz0, k);
  f4 div_z = bd_s(sxz, idx, sxz0, xe, SX) + bd_s(syz, idx, syz0, ye, SY) + bd_z(szz, idx, szz0, k);

  f4 dtr = ld4(dt_rho + m);
  f4 db  = ld4(dampb + m);

  float sv = srcxy[(size_t)b * NX * NY + (size_t)i * NY + j];
  f4 s4 = ld4(pz + k) * sv;   // AMP = 1

  st4(vx + idx, (ld4(vx + idx) + div_x * dtr) * db);
  st4(vy + idx, (ld4(vy + idx) + div_y * dtr) * db);
  st4(vz + idx, (ld4(vz + idx) + (div_z + s4) * dtr) * db);
}

// ---------------- final outputs: vz and |shear| ----------------
__global__ __launch_bounds__(256) void finalize_kernel(
    const float* __restrict__ vz, const float* __restrict__ sxy,
    const float* __restrict__ sxz, const float* __restrict__ syz,
    float* __restrict__ out)
{
  size_t q = ((size_t)blockIdx.x * blockDim.x + threadIdx.x) * 4;
  if (q >= (size_t)N4) return;
  st4(out + q, ld4(vz + q));
  f4 a = ld4(sxy + q), b2 = ld4(sxz + q), c = ld4(syz + q);
  f4 s = a * a + b2 * b2 + c * c + PHYS_EPS;
  f4 r; r.x = sqrtf(s.x); r.y = sqrtf(s.y); r.z = sqrtf(s.z); r.w = sqrtf(s.w);
  st4(out + (size_t)N4 + q, r);
}

// ---------------- host orchestration ----------------
extern "C" void kernel_launch(void* const* d_in, const int* in_sizes, int n_in,
                              void* d_out, int out_size, void* d_ws, size_t ws_size,
                              hipStream_t stream)
{
  const float* traj = (const float*)d_in[0];
  const float* gx   = (const float*)d_in[1];
  const float* gy   = (const float*)d_in[2];
  const float* gz   = (const float*)d_in[3];
  const float* rho  = (const float*)d_in[4];
  const float* mu   = (const float*)d_in[5];
  const float* lam  = (const float*)d_in[6];
  const float* eta  = (const float*)d_in[7];
  const float* damp = (const float*)d_in[8];

  float* ws  = (float*)d_ws;
  float* vx  = ws + 0 * (size_t)N4;
  float* vy  = ws + 1 * (size_t)N4;
  float* vz  = ws + 2 * (size_t)N4;
  float* sxx = ws + 3 * (size_t)N4;
  float* syy = ws + 4 * (size_t)N4;
  float* szz = ws + 5 * (size_t)N4;
  float* sxy = ws + 6 * (size_t)N4;
  float* sxz = ws + 7 * (size_t)N4;
  float* syz = ws + 8 * (size_t)N4;
  float* c_l2m  = ws + 9 * (size_t)N4;
  float* c_lam  = c_l2m + (size_t)N3;
  float* c_me2  = c_lam + (size_t)N3;
  float* dt_rho = c_me2 + (size_t)N3;
  float* dampb  = dt_rho + (size_t)N3;
  float* px  = dampb + (size_t)N3;            // T*B*NX = 3072
  float* py  = px + (size_t)T * B * NX;       // 3072
  float* pz  = py + (size_t)T * B * NY;       // 64 (16B aligned)
  float* srctab = pz + NZ;                    // T*B*NX*NY = 393216

  // zero-init all 9 state fields (ws is poisoned by the harness)
  hipMemsetAsync(d_ws, 0, (size_t)9 * N4 * sizeof(float), stream);

  profiles_kernel<<<25, 256, 0, stream>>>(traj, gx, gy, gz, px, py, pz);
  matcoef_kernel<<<N3 / 4 / 256, 256, 0, stream>>>(mu, lam, eta, rho, damp,
                                                   c_l2m, c_lam, c_me2, dt_rho, dampb);
  srcxy_wmma<<<(T * B * 64 * 32) / 256, 256, 0, stream>>>(px, py, srctab);

  dim3 blk(16, 16, 1);
  dim3 grd(NY / 16, NX, B);
  for (int t = 0; t < T; ++t) {
    stress_step<<<grd, blk, 0, stream>>>(vx, vy, vz, sxx, syy, szz, sxy, sxz, syz,
                                         c_l2m, c_lam, c_me2);
    velocity_step<<<grd, blk, 0, stream>>>(vx, vy, vz, sxx, syy, szz, sxy, sxz, syz,
                                           dt_rho, dampb,
                                           srctab + (size_t)t * B * NX * NY, pz);
  }
  finalize_kernel<<<N4 / 4 / 256, 256, 0, stream>>>(vz, sxy, sxz, syz, (float*)d_out);
}